// CRF_BiLSTM_5875515261306
// MI455X (gfx1250) — compile-verified
//
#include <hip/hip_runtime.h>
#include <hip/hip_bf16.h>
#include <math.h>

// CRF forward (logz) + Viterbi for B=512, T=1024, K=48 on gfx1250 (MI455X).
// Forward pass runs in the scaled-exp domain so the logsumexp recurrence
// becomes a [16x48]x[48x48] fp32 matmul per step -> V_WMMA_F32_16X16X4_F32,
// with per-step feature tiles streamed into LDS by the Tensor Data Mover
// (double-buffered, s_wait_tensorcnt). Viterbi is max-plus (VALU) with the
// full backpointer matrix resident in LDS (320KB/WGP) so the serial
// backtrace never touches global memory.

#define B_ 512
#define T_ 1024
#define K_ 48
#define START_ 46
#define END_ 47
#define NEG_ (-10000.0f)

typedef float v2f __attribute__((ext_vector_type(2)));
typedef float v8f __attribute__((ext_vector_type(8)));
typedef unsigned int u32x4 __attribute__((ext_vector_type(4)));
typedef int i32x4 __attribute__((ext_vector_type(4)));
typedef int i32x8 __attribute__((ext_vector_type(8)));

#if defined(__has_builtin)
#if __has_builtin(__builtin_amdgcn_tensor_load_to_lds) && __has_builtin(__builtin_amdgcn_s_wait_tensorcnt)
#define USE_TDM 1
#endif
#endif

// ---------------------------------------------------------------------------
// Kernel 1: mask transitions, build exp-domain B matrix (transposed so that
// WMMA computes D[b,next] = sum_prev A[b,prev] * exp(trans[next,prev])),
// and exp(trans[END,:]) for the final reduction.
// ---------------------------------------------------------------------------
__global__ void crf_prep(const float* __restrict__ trans,
                         float* __restrict__ transM,   // [K,K] masked (log domain)
                         float* __restrict__ Bmat,     // [prev][next] = exp(transM[next][prev])
                         float* __restrict__ EendExp)  // [K] = exp(transM[END][prev])
{
  for (int i = threadIdx.x; i < K_ * K_; i += blockDim.x) {
    int n = i / K_, p = i % K_;
    float v = trans[i];
    if (n == START_) v = NEG_;   // no transition into START
    if (p == END_)   v = NEG_;   // no transition out of END
    transM[i] = v;
    Bmat[p * K_ + n] = __expf(v);            // exp(-1e4) underflows to 0 == hard mask
    if (n == END_) EendExp[p] = __expf(v);
  }
}

// ---------------------------------------------------------------------------
// Async staging of one [16 batch x 48 f32] feature tile into LDS.
// TDM path: one tensor_load_to_lds with a 2D D# (row stride = T*K elements).
// Fallback: 6 global_load_async_to_lds_b128 per lane-group via inline asm.
// ---------------------------------------------------------------------------
__device__ __forceinline__ void feat_tile_issue(const float* __restrict__ gRow0,
                                                unsigned ldsOff, int lane) {
#ifdef USE_TDM
  (void)lane;
  unsigned long long ga = (unsigned long long)(uintptr_t)gRow0;
  u32x4 g0;
  g0.x = 1u;                                               // count=1, user descriptor
  g0.y = ldsOff;                                           // lds_addr (bytes)
  g0.z = (unsigned)(ga & 0xFFFFFFFFu);                     // global_addr[31:0]
  g0.w = (unsigned)((ga >> 32) & 0x01FFFFFFu) | (2u << 30);// global_addr[56:32] | type=2
  i32x8 g1;
  g1[0] = 0x20000;                  // workgroup_mask=0, data_size=2 (4 bytes)
  g1[1] = (int)(48u << 16);         // tensor_dim0 = 48 (bits [79:48], low half)
  g1[2] = (int)(16u << 16);         // tensor_dim1 = 16 (bits [111:80], low half)
  g1[3] = (int)(48u << 16);         // tile_dim0 = 48   (bits [127:112])
  g1[4] = 16;                       // tile_dim1 = 16   (bits [143:128])
  g1[5] = (int)(T_ * K_);           // tensor_dim0_stride = 49152 elements
  g1[6] = 0;
  g1[7] = 0;
  i32x4 z4 = {0, 0, 0, 0};          // groups 2/3 unused (<=2D tensor)
  i32x8 z8 = {0, 0, 0, 0, 0, 0, 0, 0};
  __builtin_amdgcn_tensor_load_to_lds(g0, g1, z4, z4, z8, 0);
#else
  // 192 x 16B transfers, 6 per lane: j = lane + 32*i ; batch m = j/12, chunk c = j%12
#pragma unroll
  for (int i = 0; i < 6; ++i) {
    int j = lane + 32 * i;
    int m = j / 12, c = j % 12;
    unsigned long long gaddr =
        (unsigned long long)(uintptr_t)(gRow0 + (size_t)m * T_ * K_) + (unsigned)(c * 16);
    unsigned dst = ldsOff + (unsigned)(m * 192 + c * 16);
    asm volatile("global_load_async_to_lds_b128 %0, %1, off"
                 :: "v"(dst), "v"(gaddr) : "memory");
  }
#endif
}

__device__ __forceinline__ void feat_tile_wait_prev_inflight1() {
#ifdef USE_TDM
  __builtin_amdgcn_s_wait_tensorcnt(1);   // previous tile done, current may fly
#else
  asm volatile("s_wait_asynccnt 6" ::: "memory");  // 6 loads per tile
#endif
  asm volatile("" ::: "memory");
}

__device__ __forceinline__ void feat_tile_wait_all() {
#ifdef USE_TDM
  __builtin_amdgcn_s_wait_tensorcnt(0);
#else
  asm volatile("s_wait_asynccnt 0" ::: "memory");
#endif
  asm volatile("" ::: "memory");
}

// ---------------------------------------------------------------------------
// Kernel 2: forward logz. One wave (32 threads) per 16-batch tile.
// Per step: 3 n-tiles x 12 chained V_WMMA_F32_16X16X4_F32, exp(feat) multiply
// + max-renormalize via LDS, log(scale) accumulated. Feats double-buffered
// in LDS by the TDM.
// ---------------------------------------------------------------------------
__global__ __launch_bounds__(32) void crf_forward_logz(
    const float* __restrict__ feats,    // [B,T,K]
    const float* __restrict__ Bmat,     // [prev][next]
    const float* __restrict__ EendExp,  // [K]
    float* __restrict__ logz)           // [B]
{
  __shared__ float aLds[16 * 50];                 // alpha tile, pitch 50 (8B-aligned pairs)
  __shared__ __align__(16) float fLds[2][16 * K_];// double-buffered feat tiles (pitch 48)

  const int lane = threadIdx.x;         // 0..31, wave32
  const int hh = lane >> 4;             // lane half (selects k/m sub-block)
  const int mn = lane & 15;             // m (A rows) or n (B/C cols)
  const int b0 = blockIdx.x * 16;

  const float* tileRow0 = feats + (size_t)b0 * T_ * K_;  // batch b0, t=0
  const unsigned fOff0 = (unsigned)(uintptr_t)&fLds[0][0];
  const unsigned fOff1 = (unsigned)(uintptr_t)&fLds[1][0];

  // kick off tile t=0
  feat_tile_issue(tileRow0, fOff0, lane);

  // ---- Preload all 36 B fragments (resident in VGPRs for the whole scan) --
  // f32 B 4x16 fragment: lanes 0-15 hold K=0(v0),1(v1); lanes 16-31 K=2,3; N=lane%16.
  v2f Bf[12][3];
#pragma unroll
  for (int kt = 0; kt < 12; ++kt)
#pragma unroll
    for (int nt = 0; nt < 3; ++nt) {
      int k = 4 * kt + 2 * hh;
      int n = nt * 16 + mn;
      Bf[kt][nt].x = Bmat[k * K_ + n];
      Bf[kt][nt].y = Bmat[(k + 1) * K_ + n];
    }

  // ---- init alpha: exp-domain -> 1 at START, 0 elsewhere ------------------
  for (int i = lane; i < 16 * 50; i += 32) aLds[i] = 0.0f;
  __syncthreads();
  if (lane < 16) aLds[lane * 50 + START_] = 1.0f;
  __syncthreads();

  // lane pairing for the rescale phase: lane = 2*m + h handles batch m, 24 cols
  const int m_d = lane >> 1, h_d = lane & 1;
  float logScale = 0.0f;

  v2f Af[12];
  for (int t = 0; t < T_; ++t) {
    // stream tile t+1 while computing tile t; wait for tile t
    if (t + 1 < T_) {
      feat_tile_issue(tileRow0 + (size_t)(t + 1) * K_, ((t + 1) & 1) ? fOff1 : fOff0, lane);
      feat_tile_wait_prev_inflight1();
    } else {
      feat_tile_wait_all();
    }
    const float* f = &fLds[t & 1][m_d * K_ + h_d * 24];

    // ---- build A fragments from LDS alpha (16x4 per k-tile) ---------------
    // f32 A 16x4 fragment: lanes 0-15 hold K=0,1; lanes 16-31 K=2,3; M=lane%16.
#pragma unroll
    for (int kt = 0; kt < 12; ++kt) {
      int k = 4 * kt + 2 * hh;
      Af[kt] = *(const v2f*)&aLds[mn * 50 + k];   // ds_load_b64, 8B aligned
    }

    // ---- D = A x B : 36 WMMAs, 3 independent 12-deep chains ---------------
    v8f acc0 = {}, acc1 = {}, acc2 = {};
#pragma unroll
    for (int kt = 0; kt < 12; ++kt) {
      acc0 = __builtin_amdgcn_wmma_f32_16x16x4_f32(false, Af[kt], false, Bf[kt][0],
                                                   (short)0, acc0, false, false);
      acc1 = __builtin_amdgcn_wmma_f32_16x16x4_f32(false, Af[kt], false, Bf[kt][1],
                                                   (short)0, acc1, false, false);
      acc2 = __builtin_amdgcn_wmma_f32_16x16x4_f32(false, Af[kt], false, Bf[kt][2],
                                                   (short)0, acc2, false, false);
    }

    // ---- scatter D back to LDS (C layout: n=lane%16, m=row+8*half) --------
#pragma unroll
    for (int r = 0; r < 8; ++r) {
      int m = r + 8 * hh;
      aLds[m * 50 + 0 * 16 + mn] = acc0[r];
      aLds[m * 50 + 1 * 16 + mn] = acc1[r];
      aLds[m * 50 + 2 * 16 + mn] = acc2[r];
    }
    __syncthreads();

    // ---- multiply exp(feat), per-batch max, renormalize -------------------
    float vbuf[24];
    float mx = 0.0f;
#pragma unroll
    for (int i = 0; i < 24; ++i) {
      float val = aLds[m_d * 50 + h_d * 24 + i] * __expf(f[i]);
      vbuf[i] = val;
      mx = fmaxf(mx, val);
    }
    mx = fmaxf(mx, __shfl_xor(mx, 1, 32));   // combine lane pair
    logScale += __logf(mx);
    float inv = 1.0f / mx;
#pragma unroll
    for (int i = 0; i < 24; ++i) aLds[m_d * 50 + h_d * 24 + i] = vbuf[i] * inv;
    __syncthreads();
  }

  // ---- final: logz = sum(log scales) + log(sum alpha * exp(trans[END])) ---
  float s = 0.0f;
#pragma unroll
  for (int i = 0; i < 24; ++i)
    s += aLds[m_d * 50 + h_d * 24 + i] * EendExp[h_d * 24 + i];
  s += __shfl_xor(s, 1, 32);
  if (h_d == 0) logz[b0 + m_d] = logScale + __logf(s);
}

// ---------------------------------------------------------------------------
// Kernel 3: Viterbi. 4 waves / block, 1 wave per batch. Backpointers live in
// LDS (T*K u8 = 48KB per wave; 320KB/WGP makes this possible on CDNA5) so the
// serial backtrace runs entirely at LDS latency.
// ---------------------------------------------------------------------------
__global__ __launch_bounds__(128) void crf_viterbi(
    const float* __restrict__ feats,   // [B,T,K]
    const float* __restrict__ transM,  // [K,K] masked (log domain)
    float* __restrict__ scores,        // [B]
    float* __restrict__ paths)         // [B,T] (written as float)
{
  extern __shared__ char smem[];
  float* tr = (float*)smem;                                  // [48][49]: pitch 49 -> conflict-free
  unsigned char* bpAll = (unsigned char*)(tr + K_ * 49);     // 4 * T_*K_ backpointers
  float* vAll = (float*)(bpAll + 4 * T_ * K_);               // 4 * 64 state vectors

  const int tid = threadIdx.x;
  const int wave = tid >> 5, lane = tid & 31;
  const int b = blockIdx.x * 4 + wave;

  for (int i = tid; i < K_ * K_; i += 128) tr[(i / K_) * 49 + (i % K_)] = transM[i];
  __syncthreads();

  unsigned char* bp = bpAll + wave * T_ * K_;
  float* v = vAll + wave * 64;
  const float* fb = feats + (size_t)b * T_ * K_;

  // init: 0 at START, NEG elsewhere
  v[lane] = (lane == START_) ? 0.0f : NEG_;
  if (lane < 16) v[32 + lane] = ((32 + lane) == START_) ? 0.0f : NEG_;

  const int next1 = 32 + lane;                 // second unit, lanes 0..15 only
  for (int t = 0; t < T_; ++t) {
    __builtin_prefetch(fb + (size_t)(t + 1) * K_ + lane, 0, 3);  // near-cache prefetch
    float m0 = -3.0e38f, m1 = -3.0e38f;
    int a0 = 0, a1 = 0;
#pragma unroll 4
    for (int p = 0; p < K_; ++p) {
      float vp = v[p];                         // LDS broadcast, same addr all lanes
      float s0 = vp + tr[lane * 49 + p];       // pitch-49: conflict-free across lanes
      if (s0 > m0) { m0 = s0; a0 = p; }        // strict > == first-occurrence argmax
      if (lane < 16) {
        float s1 = vp + tr[next1 * 49 + p];
        if (s1 > m1) { m1 = s1; a1 = p; }
      }
    }
    // all reads of v[] above precede these writes in wave program order
    bp[t * K_ + lane] = (unsigned char)a0;
    v[lane] = m0 + fb[(size_t)t * K_ + lane];
    if (lane < 16) {
      bp[t * K_ + next1] = (unsigned char)a1;
      v[next1] = m1 + fb[(size_t)t * K_ + next1];
    }
  }

  // terminal scores: term = v + trans[END]
  v[lane] = v[lane] + tr[END_ * 49 + lane];
  if (lane < 16) v[next1] = v[next1] + tr[END_ * 49 + next1];

  if (lane == 0) {
    float best = v[0];
    int last = 0;
    for (int j = 1; j < K_; ++j)
      if (v[j] > best) { best = v[j]; last = j; }
    scores[b] = best;
    // backtrace: path[t]=carry; carry=bp[t][carry], t = T-1..0 (all in LDS)
    int tag = last;
    float* pb = paths + (size_t)b * T_;
    for (int t = T_ - 1; t >= 0; --t) {
      pb[t] = (float)tag;
      tag = bp[t * K_ + tag];
    }
  }
}

// ---------------------------------------------------------------------------
extern "C" void kernel_launch(void* const* d_in, const int* in_sizes, int n_in,
                              void* d_out, int out_size, void* d_ws, size_t ws_size,
                              hipStream_t stream) {
  (void)in_sizes; (void)n_in; (void)out_size; (void)ws_size;
  const float* feats       = (const float*)d_in[0];   // [B,T,K]
  const float* transitions = (const float*)d_in[1];   // [K,K]

  float* ws      = (float*)d_ws;
  float* transM  = ws;                // 2304 floats
  float* Bmat    = ws + K_ * K_;      // 2304 floats
  float* EendExp = ws + 2 * K_ * K_;  // 48 floats

  float* out    = (float*)d_out;
  float* logz   = out;                // [512]
  float* scores = out + B_;           // [512]
  float* paths  = out + 2 * B_;       // [512*1024]

  crf_prep<<<1, 256, 0, stream>>>(transitions, transM, Bmat, EendExp);
  crf_forward_logz<<<B_ / 16, 32, 0, stream>>>(feats, Bmat, EendExp, logz);

  const size_t smem = (size_t)K_ * 49 * sizeof(float)      // trans tile
                    + (size_t)4 * T_ * K_                  // 4 waves of backpointers
                    + (size_t)4 * 64 * sizeof(float);      // 4 state vectors
  crf_viterbi<<<B_ / 4, 128, smem, stream>>>(feats, transM, scores, paths);
}